// CNNEncoder_61830349193581
// MI455X (gfx1250) — compile-verified
//
#include <hip/hip_runtime.h>

typedef float v2f __attribute__((ext_vector_type(2)));
typedef float v4f __attribute__((ext_vector_type(4)));
typedef float v8f __attribute__((ext_vector_type(8)));

// ---------------------------------------------------------------------------
// Prep kernel: fold the four stride-3 convolutions into a single 64x64 matrix
// C (row n = output feature, col k = embedded-vector index), plus padded Wp
// and the two bias vectors. One 64-thread block; thread n builds row n.
// ws layout (floats): [0,4096) B1 (n*64+k), [4096,8192) B2 (n*64+k = Wp[n][k]),
//                     [8192,8256) bias1, [8256,8320) bias2
// ---------------------------------------------------------------------------
__global__ __launch_bounds__(64) void cnn_prep_kernel(
    const float* __restrict__ w1, const float* __restrict__ b1,
    const float* __restrict__ w2, const float* __restrict__ b2,
    const float* __restrict__ w3, const float* __restrict__ b3,
    const float* __restrict__ w4, const float* __restrict__ b4,
    const float* __restrict__ Wp, const float* __restrict__ bp,
    float* __restrict__ ws)
{
    const int n = threadIdx.x;            // 0..63
    float row[64];
#pragma unroll
    for (int k = 0; k < 64; ++k) row[k] = 0.0f;
    float bias = 0.0f;

    if (n < 50) {
        const float* w; float bb; int off, Kw, pad;
        if (n < 14)      { w = w1; bb = b1[0]; off = 0;  Kw = 21; pad = 10; }
        else if (n < 27) { w = w2; bb = b2[0]; off = 14; Kw = 24; pad = 12; }
        else if (n < 39) { w = w3; bb = b3[0]; off = 27; Kw = 27; pad = 13; }
        else             { w = w4; bb = b4[0]; off = 39; Kw = 30; pad = 15; }
        const int p = n - off;            // local output position, stride 3
        for (int t = 0; t < Kw; ++t) {
            const int i = p * 3 + t - pad;           // embedded index
            if (i >= 0 && i < 60) row[i] += w[t];
        }
        bias = bb;
    }

    float* B1  = ws;
    float* B2  = ws + 4096;
    float* Bi1 = ws + 8192;
    float* Bi2 = ws + 8256;
    for (int k = 0; k < 64; ++k) B1[n * 64 + k] = row[k];
    Bi1[n] = bias;
    for (int k = 0; k < 64; ++k)
        B2[n * 64 + k] = (n < 50 && k < 50) ? Wp[n * 50 + k] : 0.0f;
    Bi2[n] = (n < 50) ? bp[n] : 0.0f;
}

// ---------------------------------------------------------------------------
// Main kernel: 128 threads = 4 waves; each wave owns 16-row tiles via a
// grid-stride loop (amortizes the one-time LDS setup over many tiles).
//   GEMM1: acc = E[16x60] x C^T      (15 wmma k-steps x 4 n-tiles)
//   relu(+bias1) -> H in wave-private LDS (cols 50..51 are exact zeros)
//   GEMM2: acc2 = H[16x52] x Wp^T    (13 wmma k-steps x 4 n-tiles)
//   output staged dense in LDS, then contiguous b128 global stores.
// LDS rows padded to stride 66 floats to break stride-64 bank conflicts.
// ---------------------------------------------------------------------------
#define BSTRIDE 66
#define ESTRIDE 66

__global__ __launch_bounds__(128) void cnn_main_kernel(
    const int* __restrict__ src, const float* __restrict__ emb,
    const float* __restrict__ ws, float* __restrict__ out, int batch)
{
    __shared__ float sB1[64 * BSTRIDE];     // conv matrix, [n][k]
    __shared__ float sB2[64 * BSTRIDE];     // Wp,         [n][k]
    __shared__ float sBias1[64];
    __shared__ float sBias2[64];
    __shared__ float sEmb[3072];            // 1000 x 3 embedding table
    __shared__ float sE[4][16 * ESTRIDE];   // per-wave E / H / out-staging tile

    const int tid = threadIdx.x;

    // One-time cooperative loads into LDS.
    for (int i = tid; i < 4096; i += 128) {
        const int n = i >> 6, k = i & 63;
        sB1[n * BSTRIDE + k] = ws[i];
        sB2[n * BSTRIDE + k] = ws[4096 + i];
    }
    if (tid < 64) { sBias1[tid] = ws[8192 + tid]; sBias2[tid] = ws[8256 + tid]; }
    for (int i = tid; i < 3000; i += 128) sEmb[i] = emb[i];
    __syncthreads();

    const int lane = tid & 31;
    const int wave = tid >> 5;
    const int lh   = lane >> 4;        // lane half: 0 or 1
    const int ln   = lane & 15;        // N / M position within tile
    float* eT = &sE[wave][0];

    const int numTiles = (batch + 15) >> 4;
    const int tileStep = gridDim.x * 4;

    for (int tile = blockIdx.x * 4 + wave; tile < numTiles; tile += tileStep) {
        const int rowBase = tile * 16;

        // Prefetch next tile's src slab (emits global_prefetch_b8).
        {
            const int nt2 = tile + tileStep;
            if (nt2 < numTiles) {
                const char* p = (const char*)(src) +
                                (size_t)nt2 * 16 * 20 * sizeof(int) +
                                (size_t)lane * 40;
                __builtin_prefetch(p, 0, 1);
            }
        }

        // ---- Build E tile: lane L handles row L/2, tokens (L&1)*10..+9 ----
        {
            const int m  = lane >> 1;
            const int t0 = (lane & 1) * 10;
            int r = rowBase + m; if (r > batch - 1) r = batch - 1;
            const int* srow = src + (size_t)r * 20;
#pragma unroll
            for (int t = 0; t < 10; ++t) {
                const int tok = srow[t0 + t];
                const int c   = (t0 + t) * 3;
                eT[m * ESTRIDE + c + 0] = sEmb[tok * 3 + 0];
                eT[m * ESTRIDE + c + 1] = sEmb[tok * 3 + 1];
                eT[m * ESTRIDE + c + 2] = sEmb[tok * 3 + 2];
            }
        }
        // Tile is wave-private: in-wave DS ordering suffices (no barrier).

        // ---- GEMM1: acc = E x C^T over K=60 (15 wmma steps) ----
        v8f acc[4] = {};
#pragma unroll
        for (int s = 0; s < 15; ++s) {
            const v2f a = *(const v2f*)&eT[ln * ESTRIDE + 4 * s + 2 * lh];
#pragma unroll
            for (int nt = 0; nt < 4; ++nt) {
                const int n = nt * 16 + ln;
                const v2f b = *(const v2f*)&sB1[n * BSTRIDE + 4 * s + 2 * lh];
                acc[nt] = __builtin_amdgcn_wmma_f32_16x16x4_f32(
                    false, a, false, b, (short)0, acc[nt], false, false);
            }
        }

        // ---- bias + ReLU, stage H (cols 0..51) into eT ----
#pragma unroll
        for (int nt = 0; nt < 4; ++nt) {
            const int n = nt * 16 + ln;
            if (nt < 3 || ln < 4) {              // only K<52 is read by GEMM2
                const float bv = sBias1[n];
#pragma unroll
                for (int v = 0; v < 8; ++v) {
                    const float h = fmaxf(acc[nt][v] + bv, 0.0f);
                    eT[(v + 8 * lh) * ESTRIDE + n] = h;
                }
            }
        }

        // ---- GEMM2: acc2 = H x Wp^T over K=52 (13 wmma steps) ----
        v8f acc2[4] = {};
#pragma unroll
        for (int s = 0; s < 13; ++s) {
            const v2f a = *(const v2f*)&eT[ln * ESTRIDE + 4 * s + 2 * lh];
#pragma unroll
            for (int nt = 0; nt < 4; ++nt) {
                const int n = nt * 16 + ln;
                const v2f b = *(const v2f*)&sB2[n * BSTRIDE + 4 * s + 2 * lh];
                acc2[nt] = __builtin_amdgcn_wmma_f32_16x16x4_f32(
                    false, a, false, b, (short)0, acc2[nt], false, false);
            }
        }

        // ---- bias + store ----
        if (rowBase + 16 <= batch) {
            // Fast path: tile output is one contiguous 3200B span of `out`.
            // Stage dense 16x50 block in LDS, then aligned b128 stores.
            float* oT = eT;                      // reuse (raw, stride 50)
#pragma unroll
            for (int nt = 0; nt < 4; ++nt) {
                const int n = nt * 16 + ln;
                if (n < 50) {
                    const float bv = sBias2[n];
#pragma unroll
                    for (int v = 0; v < 8; ++v)
                        oT[(v + 8 * lh) * 50 + n] = acc2[nt][v] + bv;
                }
            }
            float* obase = out + (size_t)rowBase * 50;   // 16B aligned
            for (int i = lane; i < 200; i += 32) {
                const v4f d = *(const v4f*)&oT[i * 4];
                *(v4f*)&obase[i * 4] = d;                // global_store_b128
            }
        } else {
            // Edge tile: guarded scalar stores.
#pragma unroll
            for (int nt = 0; nt < 4; ++nt) {
                const int n = nt * 16 + ln;
                if (n < 50) {
                    const float bv = sBias2[n];
#pragma unroll
                    for (int v = 0; v < 8; ++v) {
                        const int row = rowBase + v + 8 * lh;
                        if (row < batch)
                            out[(size_t)row * 50 + n] = acc2[nt][v] + bv;
                    }
                }
            }
        }
    }
}

// ---------------------------------------------------------------------------
extern "C" void kernel_launch(void* const* d_in, const int* in_sizes, int n_in,
                              void* d_out, int out_size, void* d_ws, size_t ws_size,
                              hipStream_t stream) {
    const int*   src = (const int*)  d_in[0];
    const float* emb = (const float*)d_in[1];
    const float* w1  = (const float*)d_in[2];
    const float* b1  = (const float*)d_in[3];
    const float* w2  = (const float*)d_in[4];
    const float* b2  = (const float*)d_in[5];
    const float* w3  = (const float*)d_in[6];
    const float* b3  = (const float*)d_in[7];
    const float* w4  = (const float*)d_in[8];
    const float* b4  = (const float*)d_in[9];
    const float* Wp  = (const float*)d_in[10];
    const float* bp  = (const float*)d_in[11];

    float* ws  = (float*)d_ws;
    float* out = (float*)d_out;
    const int batch = in_sizes[0] / 20;

    cnn_prep_kernel<<<1, 64, 0, stream>>>(w1, b1, w2, b2, w3, b3, w4, b4,
                                          Wp, bp, ws);

    const int numTiles = (batch + 15) / 16;          // 16-row wave tiles
    int blocks = (numTiles + 3) / 4;                 // 4 waves per block
    if (blocks > 2048) blocks = 2048;                // grid-stride loop covers rest
    if (blocks < 1) blocks = 1;
    cnn_main_kernel<<<blocks, 128, 0, stream>>>(src, emb, ws, out, batch);
}